// PillarMaxPooling_25958782337394
// MI455X (gfx1250) — compile-verified
//
#include <hip/hip_runtime.h>

typedef __attribute__((ext_vector_type(2))) float v2f;
typedef __attribute__((ext_vector_type(8))) float v8f;

#define C_IN   10
#define C_OUT  64
#define K_PAD  12   // 3 WMMA k-steps of 4

// ---------------------------------------------------------------------------
// Fold BatchNorm (eval) into the linear weights:  W' = W * s,  b' = beta - mu*s
// with s = gamma * rsqrt(var + eps).  Pad K rows 10,11 with zeros (these pads
// make the A-fragment contents for K>=10 don't-cares).
// ---------------------------------------------------------------------------
__global__ void fold_bn_kernel(const float* __restrict__ W,
                               const float* __restrict__ gamma,
                               const float* __restrict__ beta,
                               const float* __restrict__ run_mean,
                               const float* __restrict__ run_var,
                               float* __restrict__ Wp,     // [12][64]
                               float* __restrict__ bias) { // [64]
    int c = threadIdx.x;
    if (c >= C_OUT) return;
    float s = gamma[c] * rsqrtf(run_var[c] + 1e-3f);
#pragma unroll
    for (int k = 0; k < C_IN; ++k)
        Wp[k * C_OUT + c] = W[k * C_OUT + c] * s;
    Wp[10 * C_OUT + c] = 0.0f;
    Wp[11 * C_OUT + c] = 0.0f;
    bias[c] = beta[c] - run_mean[c] * s;
}

// ---------------------------------------------------------------------------
// Zero the pooled output (ReLU >= 0, so max-with-0 init == segment_max with
// empty pillars replaced by 0).  out_size is always a multiple of 64.
// ---------------------------------------------------------------------------
__global__ void zero_kernel(float4* __restrict__ o4, long long n4) {
    long long stride = (long long)gridDim.x * blockDim.x;
    for (long long i = (long long)blockIdx.x * blockDim.x + threadIdx.x; i < n4; i += stride)
        o4[i] = float4{0.f, 0.f, 0.f, 0.f};
}

// ---------------------------------------------------------------------------
// Main kernel: each wave32 owns a tile of 16 points.  fp32 WMMA 16x16x4,
// 3 accumulating k-steps (K=10 padded to 12), 4 N-tiles (64 channels).
// Branchless epilogue: +bias, ReLU, scatter-max via native u32 atomic max
// (nonneg f32 bit pattern order == unsigned order).
//
// A layout (ISA 32-bit A 16x4): lanes 0-15 hold K={0,1}, lanes 16-31 K={2,3}
// B layout (4x16):              lanes 0-15 hold K={0,1}, lanes 16-31 K={2,3}
// D layout (16x16 f32):         VGPR r, lanes 0-15 -> M=r, lanes 16-31 -> M=r+8
// ---------------------------------------------------------------------------
__global__ void pillar_wmma_kernel(const float* __restrict__ pts,      // [N][10]
                                   const int*   __restrict__ idx,      // [N]
                                   const float* __restrict__ Wp,       // [12][64]
                                   const float* __restrict__ bias,     // [64]
                                   unsigned int* __restrict__ out,     // [M][64] u32 bits
                                   int n_points, int n_tiles) {
    const int lane  = threadIdx.x & 31;
    const int half  = lane >> 4;    // 0: K pair {0,1}; 1: K pair {2,3}
    const int l16   = lane & 15;

    const int waves_per_block = blockDim.x >> 5;
    const int wave_id = blockIdx.x * waves_per_block + (threadIdx.x >> 5);
    const int n_waves = gridDim.x * waves_per_block;

    // Hoist B fragments (entire folded 12x64 weight table) into VGPRs.
    v2f bfrag[4][3];
#pragma unroll
    for (int nt = 0; nt < 4; ++nt) {
        const int col = nt * 16 + l16;
#pragma unroll
        for (int ks = 0; ks < 3; ++ks) {
            const int krow = ks * 4 + half * 2;
            v2f b;
            b.x = Wp[krow * C_OUT + col];
            b.y = Wp[(krow + 1) * C_OUT + col];
            bfrag[nt][ks] = b;
        }
    }
    float bb[4];
#pragma unroll
    for (int nt = 0; nt < 4; ++nt) bb[nt] = bias[nt * 16 + l16];

    for (int t = wave_id; t < n_tiles; t += n_waves) {
        const int base  = t * 16;
        const int ubase = __builtin_amdgcn_readfirstlane(base);  // wave-uniform

        // --- A fragments: clamped, unconditional v2f loads (8B aligned) ----
        const int p      = base + l16;
        const int pclamp = p < n_points ? p : (n_points - 1);
        const float* row = pts + (long long)pclamp * C_IN;
        const int k0 = half * 2;                         // 0 or 2
        v2f a0 = *(const v2f*)(row + k0);                // K 0..3
        v2f a1 = *(const v2f*)(row + 4 + k0);            // K 4..7
        v2f t89 = *(const v2f*)(row + 8);                // K 8..9
        v2f a2;                                          // K 10..11 are don't-care
        a2.x = half ? 0.f : t89.x;                       // (weight pad rows are 0)
        a2.y = half ? 0.f : t89.y;

        // --- pillar ids for the 16 rows: wave-uniform scalar loads ---------
        int pids[8];
#pragma unroll
        for (int r = 0; r < 8; ++r) {
            int alo = ubase + r;      alo = alo < n_points ? alo : (n_points - 1);
            int ahi = ubase + r + 8;  ahi = ahi < n_points ? ahi : (n_points - 1);
            const int lo = idx[alo];                     // uniform -> scalar path
            const int hi = idx[ahi];
            pids[r] = half ? hi : lo;                    // this lane's row r+8*half
        }

        // --- 12 WMMAs: 4 independent accumulation chains, interleaved ------
        v8f acc[4];
#pragma unroll
        for (int nt = 0; nt < 4; ++nt) {
            v8f z = {0.f, 0.f, 0.f, 0.f, 0.f, 0.f, 0.f, 0.f};
            acc[nt] = __builtin_amdgcn_wmma_f32_16x16x4_f32(false, a0, false, bfrag[nt][0],
                                                            (short)0, z, false, false);
        }
#pragma unroll
        for (int nt = 0; nt < 4; ++nt)
            acc[nt] = __builtin_amdgcn_wmma_f32_16x16x4_f32(false, a1, false, bfrag[nt][1],
                                                            (short)0, acc[nt], false, false);
#pragma unroll
        for (int nt = 0; nt < 4; ++nt)
            acc[nt] = __builtin_amdgcn_wmma_f32_16x16x4_f32(false, a2, false, bfrag[nt][2],
                                                            (short)0, acc[nt], false, false);

        // --- branchless epilogue: bias + ReLU + u32 atomic max -------------
#pragma unroll
        for (int r = 0; r < 8; ++r) {
            const int  m     = r + half * 8;             // D row this VGPR holds
            const bool vrow  = (base + m) < n_points;
            const unsigned dstb = (unsigned)pids[r] * (unsigned)C_OUT;
#pragma unroll
            for (int nt = 0; nt < 4; ++nt) {
                float h = acc[nt][r] + bb[nt];
                h = h > 0.f ? h : 0.f;                   // ReLU -> h >= 0
                h = vrow ? h : 0.f;                      // tail rows: harmless max-0
                atomicMax(out + (dstb + (unsigned)(nt * 16 + l16)),
                          __float_as_uint(h));
            }
        }
    }
}

extern "C" void kernel_launch(void* const* d_in, const int* in_sizes, int n_in,
                              void* d_out, int out_size, void* d_ws, size_t ws_size,
                              hipStream_t stream) {
    const float* pts      = (const float*)d_in[0];  // (N, 10)
    const int*   idx      = (const int*)  d_in[1];  // (N,)
    const float* W        = (const float*)d_in[2];  // (10, 64)
    const float* gamma    = (const float*)d_in[3];
    const float* beta     = (const float*)d_in[4];
    const float* run_mean = (const float*)d_in[5];
    const float* run_var  = (const float*)d_in[6];

    const int n_points = in_sizes[1];
    const int n_tiles  = (n_points + 15) / 16;

    float* Wp   = (float*)d_ws;          // 12*64 floats
    float* bias = Wp + K_PAD * C_OUT;    // 64 floats

    // 1) fold BN into weights/bias
    fold_bn_kernel<<<1, 64, 0, stream>>>(W, gamma, beta, run_mean, run_var, Wp, bias);

    // 2) zero-init pooled output (identity for max of ReLU'd values)
    {
        long long n4 = (long long)out_size / 4;   // out_size is a multiple of 64
        int threads = 256;
        long long want = (n4 + threads - 1) / threads;
        int blocks = (int)(want > 4096 ? 4096 : (want > 0 ? want : 1));
        zero_kernel<<<blocks, threads, 0, stream>>>((float4*)d_out, n4);
    }

    // 3) fused Linear+BN+ReLU via fp32 WMMA + scatter-max atomics
    {
        const int threads = 256;                  // 8 waves per block
        const int waves_per_block = threads / 32;
        int blocks = (n_tiles + waves_per_block - 1) / waves_per_block;
        if (blocks > 2048) blocks = 2048;         // grid-stride over tiles
        pillar_wmma_kernel<<<blocks, threads, 0, stream>>>(
            pts, idx, Wp, bias, (unsigned int*)d_out, n_points, n_tiles);
    }
}